// GNNModel_60833916780698
// MI455X (gfx1250) — compile-verified
//
#include <hip/hip_runtime.h>
#include <hip/hip_bf16.h>

typedef __attribute__((ext_vector_type(2))) float v2f;
typedef __attribute__((ext_vector_type(8))) float v8f;

// ---------------------------------------------------------------------------
// 1) zero workspace
// ---------------------------------------------------------------------------
__global__ void gnn_zero_f32(float* __restrict__ p, int n) {
    int i = blockIdx.x * blockDim.x + threadIdx.x;
    if (i < n) p[i] = 0.0f;
}

// ---------------------------------------------------------------------------
// 2) degree counting: deg_out[src]++, deg_in[dst]++  (float atomics, exact:
//    max possible degree 1.6M << 2^24)
// ---------------------------------------------------------------------------
__global__ void gnn_edge_degree(const int* __restrict__ src, const int* __restrict__ dst,
                                float* __restrict__ deg_out, float* __restrict__ deg_in, int nE) {
    int e = blockIdx.x * blockDim.x + threadIdx.x;
    if (e < nE) {
        atomicAdd(&deg_out[src[e]], 1.0f);
        atomicAdd(&deg_in[dst[e]], 1.0f);
    }
}

// ---------------------------------------------------------------------------
// 3) norms in-place (deg -> rsqrt(max(deg,1))) and s1 = x * norm_src
// ---------------------------------------------------------------------------
__global__ void gnn_node_norm(const float* __restrict__ x,
                              float* __restrict__ deg_out, float* __restrict__ deg_in,
                              float* __restrict__ s1, int nN) {
    int n = blockIdx.x * blockDim.x + threadIdx.x;
    if (n < nN) {
        float ns = 1.0f / sqrtf(fmaxf(deg_out[n], 1.0f));
        float nd = 1.0f / sqrtf(fmaxf(deg_in[n], 1.0f));
        deg_out[n] = ns;          // becomes norm_src
        deg_in[n]  = nd;          // becomes norm_dst
        s1[n] = x[n] * ns;
    }
}

// ---------------------------------------------------------------------------
// 4)/6) scalar scatter-add over edges: acc[dst[e]] += val[src[e]]
// ---------------------------------------------------------------------------
__global__ void gnn_edge_scatter(const int* __restrict__ src, const int* __restrict__ dst,
                                 const float* __restrict__ val, float* __restrict__ acc, int nE) {
    int e = blockIdx.x * blockDim.x + threadIdx.x;
    if (e < nE) {
        atomicAdd(&acc[dst[e]], val[src[e]]);
    }
}

// ---------------------------------------------------------------------------
// 5) per-node 1->32->1 MLP via fp32 WMMA (16x16x4).
//    a1[m] = agg1[m]*norm_dst[m].
//    A: 16x4 f32, only K=0 nonzero (lanes 0-15, VGPR0) -> D = a1[m]*W1[n] EXACT.
//    B: every slot of lane l = W1[chunk*16 + (l&15)]; dead K rows multiplied by 0.
//    Bias + ReLU + *W2 in VALU, then 16-lane butterfly reduction per half.
//    t[m] = norm_src[m] * sum_n relu(a1*W1[n]+b1[n])*W2[n].
// ---------------------------------------------------------------------------
__global__ void __launch_bounds__(256)
gnn_node_mlp_wmma(const float* __restrict__ agg1,
                  const float* __restrict__ nsrc,   // norm_src
                  const float* __restrict__ ndst,   // norm_dst
                  const float* __restrict__ W1,     // [32]
                  const float* __restrict__ b1,     // [32]
                  const float* __restrict__ W2,     // [32]
                  float* __restrict__ tout, int nN) {
    const int lane = threadIdx.x & 31;
    const int wave = threadIdx.x >> 5;
    const int gw   = blockIdx.x * (blockDim.x >> 5) + wave;
    const int base = gw * 16;
    if (base >= nN) return;              // wave-uniform: EXEC stays all-ones below

    const int col  = lane & 15;          // M for A-load; N for C/D columns
    const int half = lane >> 4;

    int nodeA = base + col;
    if (nodeA >= nN) nodeA = nN - 1;     // clamped load (no lane masking)
    float av = agg1[nodeA] * ndst[nodeA];
    float a1 = half ? 0.0f : av;         // A[M][K=0] lives in VGPR0, lanes 0-15

    v2f A;  A.x = a1;  A.y = 0.0f;       // K=1..3 slots zero
    float w1a = W1[col], w1b = W1[col + 16];
    v2f B0; B0.x = w1a; B0.y = w1a;      // column value in every K slot (K>0 dead)
    v2f B1; B1.x = w1b; B1.y = w1b;
    v8f Cz = {0.f, 0.f, 0.f, 0.f, 0.f, 0.f, 0.f, 0.f};

    // D[m][n] = a1[m] * W1[n]   (exact fp32: single nonzero product per element)
    v8f D0 = __builtin_amdgcn_wmma_f32_16x16x4_f32(false, A, false, B0, (short)0, Cz, false, false);
    v8f D1 = __builtin_amdgcn_wmma_f32_16x16x4_f32(false, A, false, B1, (short)0, Cz, false, false);

    float b1a = b1[col], b1b = b1[col + 16];
    float w2a = W2[col], w2b = W2[col + 16];

    // C/D layout: VGPR r = row (r + 8*half), column = lane&15
    float ps[8];
#pragma unroll
    for (int r = 0; r < 8; ++r) {
        float h0 = fmaxf(D0[r] + b1a, 0.0f);
        float h1 = fmaxf(D1[r] + b1b, 0.0f);
        ps[r] = h0 * w2a + h1 * w2b;
    }
    // butterfly reduce across the 16 lanes of each half (masks < 16 stay in-half)
#pragma unroll
    for (int m = 1; m < 16; m <<= 1) {
#pragma unroll
        for (int r = 0; r < 8; ++r) ps[r] += __shfl_xor(ps[r], m, 32);
    }
    if (col == 0) {
#pragma unroll
        for (int r = 0; r < 8; ++r) {
            int node = base + half * 8 + r;
            if (node < nN) tout[node] = nsrc[node] * ps[r];
        }
    }
}

// ---------------------------------------------------------------------------
// 7) out[n] = agg2[n] * norm_dst[n] + b2
// ---------------------------------------------------------------------------
__global__ void gnn_node_out(const float* __restrict__ agg2, const float* __restrict__ ndst,
                             const float* __restrict__ b2, float* __restrict__ out, int nN) {
    int n = blockIdx.x * blockDim.x + threadIdx.x;
    if (n < nN) out[n] = agg2[n] * ndst[n] + b2[0];
}

// ---------------------------------------------------------------------------
// launch
// ---------------------------------------------------------------------------
extern "C" void kernel_launch(void* const* d_in, const int* in_sizes, int n_in,
                              void* d_out, int out_size, void* d_ws, size_t ws_size,
                              hipStream_t stream) {
    const float* x   = (const float*)d_in[0];
    const int*   src = (const int*)d_in[1];
    const int*   dst = (const int*)d_in[2];
    const float* W1  = (const float*)d_in[3];
    const float* b1  = (const float*)d_in[4];
    const float* W2  = (const float*)d_in[5];
    const float* b2  = (const float*)d_in[6];
    float* out = (float*)d_out;

    const int nN = in_sizes[0];   // 100,000
    const int nE = in_sizes[1];   // 1,600,000

    float* ws      = (float*)d_ws;        // 6*nN floats = 2.4 MB
    float* deg_out = ws;                  // -> norm_src after gnn_node_norm
    float* deg_in  = ws + (size_t)nN;     // -> norm_dst after gnn_node_norm
    float* s1      = ws + (size_t)2 * nN;
    float* agg1    = ws + (size_t)3 * nN;
    float* tbuf    = ws + (size_t)4 * nN;
    float* agg2    = ws + (size_t)5 * nN;

    const int TB = 256;
    const int nzero = 6 * nN;
    gnn_zero_f32<<<(nzero + TB - 1) / TB, TB, 0, stream>>>(ws, nzero);

    gnn_edge_degree<<<(nE + TB - 1) / TB, TB, 0, stream>>>(src, dst, deg_out, deg_in, nE);
    gnn_node_norm<<<(nN + TB - 1) / TB, TB, 0, stream>>>(x, deg_out, deg_in, s1, nN);

    gnn_edge_scatter<<<(nE + TB - 1) / TB, TB, 0, stream>>>(src, dst, s1, agg1, nE);

    const int waves  = (nN + 15) / 16;        // 16 nodes per wave
    const int blocks = (waves + 7) / 8;       // 8 waves (256 threads) per block
    gnn_node_mlp_wmma<<<blocks, 256, 0, stream>>>(agg1, deg_out, deg_in, W1, b1, W2, tbuf, nN);

    gnn_edge_scatter<<<(nE + TB - 1) / TB, TB, 0, stream>>>(src, dst, tbuf, agg2, nE);
    gnn_node_out<<<(nN + TB - 1) / TB, TB, 0, stream>>>(agg2, deg_in, b2, out, nN);
}